// PSRoIPool_82394652606918
// MI455X (gfx1250) — compile-verified
//
#include <hip/hip_runtime.h>

// ---------------------------------------------------------------------------
// PSRoI pooling on MI455X (gfx1250), fp32 end-to-end via V_WMMA_F32_16X16X4_F32,
// with async global->LDS staging (GLOBAL_LOAD_ASYNC_TO_LDS_B128, ASYNCcnt).
//
// Shapes fixed by the reference setup: B=2, D=8, G=7, C=392, H=50, W=84, N=128.
// Roofline: ~13 MB features + ~1 MB masks => <1 us at 23.3 TB/s; ~0.9 GFLOP of
// f32 WMMA. Design: one block per channel, channel image staged to LDS once
// (minimum HBM traffic), 16-ROI GEMM tiles per wave.
// ---------------------------------------------------------------------------

typedef __attribute__((ext_vector_type(2))) float v2f;
typedef __attribute__((ext_vector_type(8))) float v8f;
typedef int v4i __attribute__((vector_size(16)));   // matches builtin param type

constexpr int GS  = 7;            // group size
constexpr int BD  = 2;            // batches
constexpr int DD  = 8;            // score maps
constexpr int CC  = DD * GS * GS; // 392 channels
constexpr int HH  = 50;
constexpr int WW  = 84;           // K1 = 84 = 21 * 4 (WMMA K chunks)
constexpr int K2P = 112;          // padded (b,h) rows: 2*50 -> 112 = 7 * 16

// Address-space helpers. Flat pointers into LDS carry the LDS offset in
// addr[31:0] (aperture rule), so the truncating int cast == addrspacecast.
typedef __attribute__((address_space(1))) v4i* glb_v4i_p;   // __device__ v4i*
typedef __attribute__((address_space(3))) v4i* lds_v4i_p;   // __shared__ v4i*
__device__ __forceinline__ glb_v4i_p to_glb(const void* p) {
  return (glb_v4i_p)(unsigned long long)p;
}
__device__ __forceinline__ lds_v4i_p to_lds(void* p) {
  return (lds_v4i_p)(unsigned int)(unsigned long long)p;
}

__device__ __forceinline__ void wait_async_zero() {
#if __has_builtin(__builtin_amdgcn_s_wait_asynccnt)
  __builtin_amdgcn_s_wait_asynccnt(0);
#else
  asm volatile("s_wait_asynccnt 0" ::: "memory");
#endif
}

// ---------------------------------------------------------------------------
// Prep: per-ROI 0/1 masks (batch folded into MH2) + reciprocal bin areas.
// Exactly mirrors the reference geometry: jnp.round == rintf (half-to-even).
// ---------------------------------------------------------------------------
__global__ void psroi_prep(const float* __restrict__ rois,
                           const int*   __restrict__ stride_p,
                           float* __restrict__ mw,    // [N][GS][WW]
                           float* __restrict__ mh2,   // [N][GS][K2P]
                           float* __restrict__ inva,  // [N][GS][GS]
                           int N) {
  const int n = blockIdx.x;
  if (n >= N) return;
  const float scale = 1.0f / (float)(*stride_p);
  const float* r = rois + n * 5;
  const int  bidx = (int)r[0];
  const float xs = rintf(r[1]) * scale;
  const float ys = rintf(r[2]) * scale;
  const float xe = (rintf(r[3]) + 1.0f) * scale;
  const float ye = (rintf(r[4]) + 1.0f) * scale;
  const float bw = fmaxf(xe - xs, 0.1f) / (float)GS;
  const float bh = fmaxf(ye - ys, 0.1f) / (float)GS;

  float wst[GS], wen[GS], hst[GS], hen[GS];
#pragma unroll
  for (int g = 0; g < GS; ++g) {
    wst[g] = fminf(fmaxf(floorf((float)g * bw + xs), 0.0f), (float)WW);
    wen[g] = fminf(fmaxf(ceilf(((float)g + 1.0f) * bw + xs), 0.0f), (float)WW);
    hst[g] = fminf(fmaxf(floorf((float)g * bh + ys), 0.0f), (float)HH);
    hen[g] = fminf(fmaxf(ceilf(((float)g + 1.0f) * bh + ys), 0.0f), (float)HH);
  }
  for (int e = threadIdx.x; e < GS * WW; e += blockDim.x) {
    const int g = e / WW, w = e - g * WW;
    const float fw = (float)w;
    mw[(n * GS + g) * WW + w] = (fw >= wst[g] && fw < wen[g]) ? 1.0f : 0.0f;
  }
  for (int e = threadIdx.x; e < GS * K2P; e += blockDim.x) {
    const int g = e / K2P, row = e - g * K2P;
    float v = 0.0f;
    if (row < BD * HH) {
      const int b = row / HH, h = row - b * HH;
      const float fh = (float)h;
      v = (b == bidx && fh >= hst[g] && fh < hen[g]) ? 1.0f : 0.0f;
    }
    mh2[(n * GS + g) * K2P + row] = v;
  }
  for (int e = threadIdx.x; e < GS * GS; e += blockDim.x) {
    const int gi = e / GS, gj = e - gi * GS;
    const float area = fmaxf((hen[gi] - hst[gi]) * (wen[gj] - wst[gj]), 1.0f);
    inva[n * GS * GS + e] = 1.0f / area;
  }
}

// ---------------------------------------------------------------------------
// Main: one block per channel c = (d*G+i)*G+j. Stage the 2-batch channel image
// into LDS once (async DMA path); each wave computes a 16-ROI tile:
//   Stage1 (WMMA f32 16x16x4): T[(b,h), n] = sum_w F[(b,h),w] * MW[n,w]
//   Stage2 (VALU on WMMA accum layout): pooled[n] = sum_row T[row,n]*MH2[n,row]
// ---------------------------------------------------------------------------
__global__ __launch_bounds__(256)
void psroi_wmma(const float* __restrict__ feats,
                const float* __restrict__ mw,
                const float* __restrict__ mh2,
                const float* __restrict__ inva,
                float* __restrict__ out, int N) {
  constexpr int ROWT   = K2P / 16;      // 7 WMMA row tiles
  constexpr int KC     = WW / 4;        // 21 WMMA K chunks
  constexpr int NCHUNK = BD * HH * 21;  // 2100 aligned B128 transfers
  __shared__ float lds[K2P * WW];       // 112*84*4 = 37632 B

  const int cch = blockIdx.x;           // channel 0..391, == (d*GS+gi)*GS+gj
  const int d   = cch / (GS * GS);
  const int rem = cch - d * GS * GS;
  const int gi  = rem / GS;
  const int gj  = rem - gi * GS;

#if __has_builtin(__builtin_amdgcn_global_load_async_to_lds_b128)
  // Zero the 12 pad rows (plain LDS stores), then DMA the 100 real rows
  // directly global->LDS, bypassing VGPRs. Each row is 336 B = 21 aligned
  // B128 transfers; 2100 transfers spread over 256 lanes.
  for (int e = threadIdx.x; e < (K2P - BD * HH) * WW; e += blockDim.x)
    lds[BD * HH * WW + e] = 0.0f;
  for (int q = threadIdx.x; q < NCHUNK; q += blockDim.x) {
    const int row  = q / 21;            // (b,h) row index
    const int part = q - row * 21;      // 16-byte chunk within the row
    const int b = row / HH, h = row - b * HH;
    const float* g = feats + (((size_t)b * CC + cch) * HH + h) * WW + part * 4;
    float*       l = lds + row * WW + part * 4;
    __builtin_amdgcn_global_load_async_to_lds_b128(to_glb(g), to_lds(l), 0, 0);
  }
  wait_async_zero();   // s_wait_asynccnt 0: this wave's DMAs landed in LDS
#else
  for (int e = threadIdx.x; e < K2P * WW; e += blockDim.x) {
    const int row = e / WW, w = e - row * WW;
    float v = 0.0f;
    if (row < BD * HH) {
      const int b = row / HH, h = row - b * HH;
      v = feats[(((size_t)b * CC + cch) * HH + h) * WW + w];
    }
    lds[e] = v;
  }
#endif
  __syncthreads();     // cross-wave visibility of the staged tile

  const int lane = threadIdx.x & 31;    // wave32
  const int wv   = threadIdx.x >> 5;
  const int col  = lane & 15;           // A row m, B/C column n
  const int kh   = lane >> 4;           // lane half: A/B k-phase, C row +8
  const int nwaves = blockDim.x >> 5;

  for (int tile = wv; tile * 16 < N; tile += nwaves) {
    const int n     = tile * 16 + col;
    const bool valid = (n < N);
    const int ncl   = valid ? n : (N - 1);
    const float* mwp = mw  + ((size_t)ncl * GS + gj) * WW;
    const float* mhp = mh2 + ((size_t)ncl * GS + gi) * K2P;
    __builtin_prefetch(mwp, 0, 1);      // global_prefetch_b8
    __builtin_prefetch(mhp, 0, 1);

    const v8f vzero = {0.f, 0.f, 0.f, 0.f, 0.f, 0.f, 0.f, 0.f};
    v8f acc[ROWT];
#pragma unroll
    for (int t = 0; t < ROWT; ++t) acc[t] = vzero;

    // K loop outer so each B fragment (per-ROI mask) is loaded once and reused
    // by all 7 row tiles; A fragments stream from LDS (conflict-free: 84-float
    // row stride maps 16 consecutive rows to 16 distinct banks).
    for (int kc = 0; kc < KC; ++kc) {
      const int w0 = kc * 4 + 2 * kh;   // A/B layout: k = 2*(lane/16) + v
      const v2f bfrag = *(const v2f*)(mwp + w0);
#pragma unroll
      for (int t = 0; t < ROWT; ++t) {
        const v2f afrag = *(const v2f*)(&lds[(t * 16 + col) * WW + w0]);
        acc[t] = __builtin_amdgcn_wmma_f32_16x16x4_f32(
            false, afrag, false, bfrag, (short)0, acc[t], false, false);
      }
    }

    // Stage 2: weighted column reduction directly on the C/D register layout
    // (VGPR v, lane half kh -> row t*16 + v + 8*kh; lane%16 -> column n).
    float s = 0.0f;
#pragma unroll
    for (int t = 0; t < ROWT; ++t)
#pragma unroll
      for (int v = 0; v < 8; ++v)
        s += acc[t][v] * mhp[t * 16 + v + 8 * kh];

    s += __shfl_xor(s, 16, 32);         // combine the two lane halves
    if (kh == 0 && valid) {
      out[(((size_t)n * DD + d) * GS + gi) * GS + gj] =
          s * inva[((size_t)n * GS + gi) * GS + gj];
    }
  }
}

// ---------------------------------------------------------------------------
extern "C" void kernel_launch(void* const* d_in, const int* in_sizes, int n_in,
                              void* d_out, int out_size, void* d_ws, size_t ws_size,
                              hipStream_t stream) {
  (void)n_in; (void)out_size; (void)ws_size;
  const float* rois     = (const float*)d_in[0];
  const float* feats    = (const float*)d_in[1];
  const int*   stride_p = (const int*)d_in[2];
  float* out = (float*)d_out;
  const int N = in_sizes[0] / 5;  // rois is [N,5]

  // Workspace layout (floats): MW | MH2 | 1/area  (~727 KB for N=128)
  float* ws   = (float*)d_ws;
  float* mw   = ws;
  float* mh2  = mw  + (size_t)N * GS * WW;
  float* inva = mh2 + (size_t)N * GS * K2P;

  psroi_prep<<<N, 128, 0, stream>>>(rois, stride_p, mw, mh2, inva, N);
  psroi_wmma<<<CC, 256, 0, stream>>>(feats, mw, mh2, inva, out, N);
}